// Router_30966714204216
// MI455X (gfx1250) — compile-verified
//
#include <hip/hip_runtime.h>
#include <hip/hip_bf16.h>
#include <math.h>

// ---------------------------------------------------------------------------
// Problem constants (match reference)
// ---------------------------------------------------------------------------
#define BATCH        256
#define XDIM         8192
#define CDIM         8
#define NFLAT        (XDIM * CDIM)          // 65536 per batch row
#define SEGS         4
#define SEG_FLAT     (NFLAT / SEGS)         // 16384
#define FFT_BINS     16
#define HIDDEN       64
#define NPRIM        16
#define STATS_DIM    29
#define IN_DIM       37
#define KPAD         40                      // IN_DIM padded to multiple of 4

// DFT-as-GEMM staging
#define TW_M         32                      // rows: 16 cos + 16 sin
#define CHUNK_X      512                     // x-positions per LDS chunk
#define CHUNK_FLAT   (CHUNK_X * CDIM)        // 4096 floats = 16 KB
#define NCHUNK       (XDIM / CHUNK_X)        // 16

typedef __attribute__((ext_vector_type(2))) float v2f;
typedef __attribute__((ext_vector_type(8))) float v8f;
typedef __attribute__((ext_vector_type(4))) int   v4i;

#define GLB __attribute__((address_space(1)))
#define LDS __attribute__((address_space(3)))

// order-preserving float <-> uint map (for atomic min/max on LDS)
__device__ __forceinline__ unsigned f2ord(float f) {
    unsigned u = __float_as_uint(f);
    return (u & 0x80000000u) ? ~u : (u | 0x80000000u);
}
__device__ __forceinline__ float ord2f(unsigned u) {
    return (u & 0x80000000u) ? __uint_as_float(u & 0x7fffffffu)
                             : __uint_as_float(~u);
}

__device__ __forceinline__ void async_wait0() {
#if __has_builtin(__builtin_amdgcn_s_wait_asynccnt)
    __builtin_amdgcn_s_wait_asynccnt(0);
#else
    asm volatile("s_wait_asynccnt 0x0" ::: "memory");
#endif
}

// Stage one 512x8 chunk of u into LDS (async DMA when available).
__device__ __forceinline__ void stage_chunk(const float* __restrict__ ub,
                                            float* dstLds, int chunk, int tid) {
    const float* src = ub + chunk * CHUNK_FLAT;
    #pragma unroll
    for (int j = 0; j < CHUNK_FLAT / (256 * 4); ++j) {
        const int off = (j * 256 + tid) * 4;          // float index, 16B granules
#if __has_builtin(__builtin_amdgcn_global_load_async_to_lds_b128)
        __builtin_amdgcn_global_load_async_to_lds_b128(
            (GLB v4i*)(src + off), (LDS v4i*)(dstLds + off), 0, 0);
#else
        *(float4*)(dstLds + off) = *(const float4*)(src + off);
#endif
    }
}

// ---------------------------------------------------------------------------
// Kernel 0: twiddle table, transposed for coalesced WMMA A-tile loads.
//   twidT[x][m] = cos(2*pi*m*x/8192)        for m in [0,16)
//                 sin(2*pi*(m-16)*x/8192)   for m in [16,32)
// (reference Im = -sum u*sin; sign drops out under |.|)
// ---------------------------------------------------------------------------
__global__ __launch_bounds__(256)
void router_twiddle_kernel(float* __restrict__ twidT) {
    const int idx = blockIdx.x * 256 + threadIdx.x;   // [0, 8192*32)
    const int x   = idx >> 5;
    const int m   = idx & 31;
    const int t   = ((m & 15) * x) & (XDIM - 1);
    const float ang = 6.28318530717958647692f / (float)XDIM * (float)t;
    float sn, cs;
    __sincosf(ang, &sn, &cs);
    twidT[idx] = (m < 16) ? cs : sn;
}

// ---------------------------------------------------------------------------
// Kernel 1: per-batch-row statistics + WMMA 16-bin DFT -> feature row [40]
// One block (256 threads = 8 wave32) per batch row.
//   Phase A: moments/extrema/grad/segments via coalesced b128 + LDS atomics
//   Phase B: DFT as [32 x 8192] @ [8192 x 8] GEMM on v_wmma_f32_16x16x4_f32,
//            B staged chunk-wise into LDS with async-to-LDS double buffering.
// ---------------------------------------------------------------------------
__global__ __launch_bounds__(256)
void router_stats_kernel(const float* __restrict__ u,
                         const float* __restrict__ pde,
                         const float* __restrict__ twidT,
                         float* __restrict__ featPad) {
    const int b   = blockIdx.x;
    const int tid = threadIdx.x;
    const float* __restrict__ ub = u + (size_t)b * NFLAT;

    __shared__ float    s_u[2 * CHUNK_FLAT];          // 32 KB double buffer
    __shared__ float    s_red[TW_M][16];              // Re rows 0-15, Im rows 16-31
    __shared__ float    s_segsum[SEGS];
    __shared__ float    s_segsq[SEGS];
    __shared__ float    s_grad;
    __shared__ unsigned s_min, s_max;
    __shared__ float    s_mag[FFT_BINS][CDIM];
    __shared__ float    s_stat[32];

    if (tid < SEGS) { s_segsum[tid] = 0.f; s_segsq[tid] = 0.f; }
    if (tid == 0)   { s_grad = 0.f; s_min = 0xFFFFFFFFu; s_max = 0u; }
    {
        float* r = &s_red[0][0];
        for (int i = tid; i < TW_M * 16; i += 256) r[i] = 0.f;
    }
    __syncthreads();

    // ---------------- Phase A: moments / extrema / grad / segment stats ----
    float mn =  3.402823466e38f;
    float mx = -3.402823466e38f;
    float g  = 0.f;
    #pragma unroll
    for (int s = 0; s < SEGS; ++s) {
        float ssum = 0.f, ssq = 0.f;
        #pragma unroll 4
        for (int it = 0; it < 16; ++it) {
            const int i = ((s * 16 + it) * 256 + tid) * 4;   // flat index, seg == s
            const float4 v = *(const float4*)(ub + i);
            ssum += v.x + v.y + v.z + v.w;
            ssq  += v.x * v.x + v.y * v.y + v.z * v.z + v.w * v.w;
            mn = fminf(mn, fminf(fminf(v.x, v.y), fminf(v.z, v.w)));
            mx = fmaxf(mx, fmaxf(fmaxf(v.x, v.y), fmaxf(v.z, v.w)));
            if (i >= CDIM) {  // grad terms exist only for flat index >= 8
                const float4 p = *(const float4*)(ub + i - CDIM);
                g += fabsf(v.x - p.x) + fabsf(v.y - p.y) +
                     fabsf(v.z - p.z) + fabsf(v.w - p.w);
            }
        }
        atomicAdd(&s_segsum[s], ssum);   // ds_add_f32
        atomicAdd(&s_segsq[s],  ssq);
    }
    atomicAdd(&s_grad, g);
    atomicMin(&s_min, f2ord(mn));
    atomicMax(&s_max, f2ord(mx));

    // ---------------- Phase B: DFT as WMMA GEMM ----------------------------
    // A (16x4 f32): M = bin, K = x, from twidT (coalesced row reads).
    // B (4x16 f32): K = x (within chunk, from LDS), N = channel (8 valid).
    const int lane  = tid & 31;
    const int lr    = lane & 15;
    const int khalf = lane >> 4;                  // K pair {0,1} vs {2,3}
    const int wave  = tid >> 5;
    const int lc    = lr & 7;                     // in-bounds LDS column
    const float cmask = (lr < CDIM) ? 1.f : 0.f;  // zero pad cols 8..15

    v8f accRe = {};
    v8f accIm = {};

    stage_chunk(ub, s_u, 0, tid);
    for (int ch = 0; ch < NCHUNK; ++ch) {
        async_wait0();
        __syncthreads();                          // chunk ch resident in LDS
        if (ch + 1 < NCHUNK)
            stage_chunk(ub, s_u + ((ch + 1) & 1) * CHUNK_FLAT, ch + 1, tid);
        const float* cu = s_u + (ch & 1) * CHUNK_FLAT;

        #pragma unroll 4
        for (int kt = wave; kt < CHUNK_X / 4; kt += 8) {
            const int kl = kt * 4 + khalf * 2;            // x within chunk
            const int ka = ch * CHUNK_X + kl;             // absolute x
            v2f bm, aRe, aIm;
            // unconditional in-bounds ds_load + select (branchless zero-pad)
            bm.x  = cmask * cu[(kl    ) * CDIM + lc];
            bm.y  = cmask * cu[(kl + 1) * CDIM + lc];
            aRe.x = twidT[(ka    ) * TW_M + lr];
            aRe.y = twidT[(ka + 1) * TW_M + lr];
            aIm.x = twidT[(ka    ) * TW_M + 16 + lr];
            aIm.y = twidT[(ka + 1) * TW_M + 16 + lr];
            accRe = __builtin_amdgcn_wmma_f32_16x16x4_f32(
                        false, aRe, false, bm, (short)0, accRe, false, false);
            accIm = __builtin_amdgcn_wmma_f32_16x16x4_f32(
                        false, aIm, false, bm, (short)0, accIm, false, false);
        }
        __syncthreads();                          // done reading before rewrite
    }

    // cross-wave reduction of the 16x16 D tiles (cols >= 8 are exact zeros)
    #pragma unroll
    for (int r = 0; r < 8; ++r) {
        const int m = r + khalf * 8;              // bin index
        atomicAdd(&s_red[m][lr],      accRe[r]);
        atomicAdd(&s_red[16 + m][lr], accIm[r]);
    }
    __syncthreads();

    if (tid < FFT_BINS * CDIM) {
        const int c = tid & 7, k = tid >> 3;
        const float R = s_red[k][c];
        const float I = s_red[16 + k][c];
        s_mag[k][c] = sqrtf(R * R + I * I);
    }
    __syncthreads();

    if (tid == 0) {
        float tot = 0.f, totsq = 0.f;
        #pragma unroll
        for (int s = 0; s < SEGS; ++s) { tot += s_segsum[s]; totsq += s_segsq[s]; }
        const float mean = tot / (float)NFLAT;
        const float var  = (totsq - tot * tot / (float)NFLAT) / (float)(NFLAT - 1);
        s_stat[0] = mean;
        s_stat[1] = sqrtf(fmaxf(var, 0.f));
        s_stat[2] = ord2f(s_min);
        s_stat[3] = ord2f(s_max);
        s_stat[4] = s_grad / (float)((XDIM - 1) * CDIM);
        #pragma unroll
        for (int s = 0; s < SEGS; ++s) {
            s_stat[5 + s] = s_segsum[s] / (float)SEG_FLAT;
            const float sv = (s_segsq[s] - s_segsum[s] * s_segsum[s] / (float)SEG_FLAT)
                             / (float)(SEG_FLAT - 1);
            s_stat[9 + s] = sqrtf(fmaxf(sv, 0.f));
        }
    }
    if (tid < FFT_BINS) {
        float acc = 0.f;
        #pragma unroll
        for (int c = 0; c < CDIM; ++c) acc += s_mag[tid][c];
        s_stat[13 + tid] = acc * (1.f / (float)CDIM);
    }
    __syncthreads();

    if (tid < KPAD) {
        const float v = (tid < STATS_DIM) ? s_stat[tid]
                      : (tid < IN_DIM)    ? pde[b * 8 + (tid - STATS_DIM)]
                                          : 0.f;
        featPad[b * KPAD + tid] = v;
    }
}

// ---------------------------------------------------------------------------
// Kernel 2: MLP trunk via V_WMMA_F32_16X16X4_F32 + top-2 masked softmax.
// 16 blocks (one per 16-row M tile) x 128 threads (4 wave32).
// ---------------------------------------------------------------------------
__global__ __launch_bounds__(128)
void router_mlp_kernel(const float* __restrict__ featPad,
                       const float* __restrict__ W1,
                       const float* __restrict__ b1,
                       const float* __restrict__ W2,
                       const float* __restrict__ b2,
                       float* __restrict__ out) {
    __shared__ float hTile[16][HIDDEN];   // relu(feat@W1+b1) tile
    __shared__ float lTile[16][NPRIM];    // logits tile

    const int lane  = threadIdx.x & 31;
    const int wave  = threadIdx.x >> 5;
    const int lr    = lane & 15;
    const int khalf = lane >> 4;
    const int mBase = blockIdx.x * 16;

    // ---- GEMM1: [16 x 40] @ [40 x 64] tile-slice --------------------------
    {
        v8f acc = {};
        #pragma unroll
        for (int kt = 0; kt < KPAD / 4; ++kt) {
            const int k0 = kt * 4 + khalf * 2;
            v2f a, bm;
            a.x = featPad[(mBase + lr) * KPAD + k0];
            a.y = featPad[(mBase + lr) * KPAD + k0 + 1];
            const int col = wave * 16 + lr;
            bm.x = (k0     < IN_DIM) ? W1[(k0    ) * HIDDEN + col] : 0.f;
            bm.y = (k0 + 1 < IN_DIM) ? W1[(k0 + 1) * HIDDEN + col] : 0.f;
            acc = __builtin_amdgcn_wmma_f32_16x16x4_f32(
                      false, a, false, bm, (short)0, acc, false, false);
        }
        const int col  = wave * 16 + lr;
        const float bb = b1[col];
        #pragma unroll
        for (int r = 0; r < 8; ++r) {
            const float hv = acc[r] + bb;
            hTile[r + khalf * 8][col] = hv > 0.f ? hv : 0.f;
        }
    }
    __syncthreads();

    // ---- GEMM2: [16 x 64] @ [64 x 16] -------------------------------------
    if (wave == 0) {
        v8f acc = {};
        #pragma unroll
        for (int kt = 0; kt < HIDDEN / 4; ++kt) {
            const int k0 = kt * 4 + khalf * 2;
            v2f a, bm;
            a.x  = hTile[lr][k0];
            a.y  = hTile[lr][k0 + 1];
            bm.x = W2[(k0    ) * NPRIM + lr];
            bm.y = W2[(k0 + 1) * NPRIM + lr];
            acc = __builtin_amdgcn_wmma_f32_16x16x4_f32(
                      false, a, false, bm, (short)0, acc, false, false);
        }
        const float bb = b2[lr];
        #pragma unroll
        for (int r = 0; r < 8; ++r)
            lTile[r + khalf * 8][lr] = acc[r] + bb;
    }
    __syncthreads();

    // ---- top-2 + always-on primitive 0 + masked softmax -------------------
    if (threadIdx.x < 16) {
        const int m = threadIdx.x;
        float lg[NPRIM];
        #pragma unroll
        for (int j = 0; j < NPRIM; ++j) lg[j] = lTile[m][j];

        int i0 = 0; float v0 = lg[0];
        #pragma unroll
        for (int j = 1; j < NPRIM; ++j)
            if (lg[j] > v0) { v0 = lg[j]; i0 = j; }
        int i1 = -1; float v1 = -3.402823466e38f;
        #pragma unroll
        for (int j = 0; j < NPRIM; ++j)
            if (j != i0 && lg[j] > v1) { v1 = lg[j]; i1 = j; }
        if (i0 != 0 && i1 != 0) { i1 = 0; v1 = lg[0]; }  // force ALWAYS_ON

        const float mxv = fmaxf(v0, v1);
        const float e0  = expf(v0 - mxv);
        const float e1  = expf(v1 - mxv);
        const float inv = 1.f / (e0 + e1);
        float* orow = out + (size_t)(mBase + m) * NPRIM;
        #pragma unroll
        for (int j = 0; j < NPRIM; ++j)
            orow[j] = (j == i0) ? e0 * inv : (j == i1) ? e1 * inv : 0.f;
    }
}

// ---------------------------------------------------------------------------
// Host launcher
// ---------------------------------------------------------------------------
extern "C" void kernel_launch(void* const* d_in, const int* in_sizes, int n_in,
                              void* d_out, int out_size, void* d_ws, size_t ws_size,
                              hipStream_t stream) {
    const float* u_state = (const float*)d_in[0];   // [256, 8192, 8]
    const float* pde     = (const float*)d_in[1];   // [256, 8]
    const float* W1      = (const float*)d_in[2];   // [37, 64]
    const float* b1      = (const float*)d_in[3];   // [64]
    const float* W2      = (const float*)d_in[4];   // [64, 16]
    const float* b2      = (const float*)d_in[5];   // [16]
    float*       out     = (float*)d_out;           // [256, 16]

    float* twidT   = (float*)d_ws;                  // [8192, 32]  (1 MB)
    float* featPad = twidT + XDIM * TW_M;           // [256, 40]

    router_twiddle_kernel<<<(XDIM * TW_M) / 256, 256, 0, stream>>>(twidT);
    router_stats_kernel<<<BATCH, 256, 0, stream>>>(u_state, pde, twidT, featPad);
    router_mlp_kernel<<<BATCH / 16, 128, 0, stream>>>(featPad, W1, b1, W2, b2, out);
}